// PointNet2MSG_61229053771916
// MI455X (gfx1250) — compile-verified
//
#include <hip/hip_runtime.h>
#include <math.h>

typedef float v2f __attribute__((ext_vector_type(2)));
typedef float v8f __attribute__((ext_vector_type(8)));

// ---------------------------------------------------------------- prep
__global__ void prep_kernel(const float* __restrict__ points, const float* __restrict__ ptsimg,
                            float* __restrict__ xyz, float* __restrict__ feat, float* __restrict__ sel,
                            int total, float invW, float invH)
{
    int i = blockIdx.x * blockDim.x + threadIdx.x;
    if (i >= total) return;
    xyz[i * 3 + 0] = points[i * 5 + 1];
    xyz[i * 3 + 1] = points[i * 5 + 2];
    xyz[i * 3 + 2] = points[i * 5 + 3];
    feat[i] = points[i * 5 + 4];
    sel[i * 2 + 0] = 2.f * ptsimg[i * 3 + 1] * invW - 1.f;
    sel[i * 2 + 1] = 2.f * ptsimg[i * 3 + 2] * invH - 1.f;
}

// ---------------------------------------------------------------- FPS
// one block per batch; iterative farthest-point sampling with argmax reduction
__global__ void fps_kernel(const float* __restrict__ xyz, float* __restrict__ dists,
                           int* __restrict__ fidx, int Nk, int npoint)
{
    int b = blockIdx.x;
    xyz   += (size_t)b * Nk * 3;
    dists += (size_t)b * Nk;
    fidx  += (size_t)b * npoint;
    __shared__ float s_val[32];
    __shared__ int   s_idx[32];
    __shared__ int   s_far;
    int tid = threadIdx.x;
    int nt  = blockDim.x;
    int far = 0;
    for (int it = 0; it < npoint; ++it) {
        if (tid == 0) fidx[it] = far;
        float cx = xyz[far * 3 + 0], cy = xyz[far * 3 + 1], cz = xyz[far * 3 + 2];
        float best = -1.f; int besti = 0x7fffffff;
        for (int i = tid; i < Nk; i += nt) {
            float dx = xyz[i * 3 + 0] - cx;
            float dy = xyz[i * 3 + 1] - cy;
            float dz = xyz[i * 3 + 2] - cz;
            float d  = dx * dx + dy * dy + dz * dz;
            float prev = (it == 0) ? 1e10f : dists[i];
            float nd = fminf(prev, d);
            dists[i] = nd;
            if (nd > best) { best = nd; besti = i; }
        }
        for (int off = 16; off > 0; off >>= 1) {
            float ov = __shfl_down(best, off, 32);
            int   oi = __shfl_down(besti, off, 32);
            if (ov > best || (ov == best && oi < besti)) { best = ov; besti = oi; }
        }
        int wid = tid >> 5;
        if ((tid & 31) == 0) { s_val[wid] = best; s_idx[wid] = besti; }
        __syncthreads();
        if (tid < 32) {
            int nw = nt >> 5;
            float v = (tid < nw) ? s_val[tid] : -1.f;
            int   i = (tid < nw) ? s_idx[tid] : 0x7fffffff;
            for (int off = 16; off > 0; off >>= 1) {
                float ov = __shfl_down(v, off, 32);
                int   oi = __shfl_down(i, off, 32);
                if (ov > v || (ov == v && oi < i)) { v = ov; i = oi; }
            }
            if (tid == 0) s_far = i;
        }
        __syncthreads();
        far = s_far;
    }
}

// ---------------------------------------------------------------- gather (generic C channels)
__global__ void gather_kernel(const float* __restrict__ src, const int* __restrict__ idx,
                              float* __restrict__ dst, int Nk, int np, int C, int nb)
{
    size_t gid = (size_t)blockIdx.x * blockDim.x + threadIdx.x;
    size_t total = (size_t)nb * np * C;
    if (gid >= total) return;
    int c = (int)(gid % C);
    size_t pj = gid / C;
    int j = (int)(pj % np);
    int b = (int)(pj / np);
    dst[gid] = src[((size_t)b * Nk + idx[(size_t)b * np + j]) * C + c];
}

// ---------------------------------------------------------------- ball query (wave per center, order-preserving compaction)
__global__ void ball_query_kernel(const float* __restrict__ xyz, const float* __restrict__ nxyz,
                                  int* __restrict__ idx, int Nk, int np, float r2, int ns, int nb)
{
    int gw   = blockIdx.x * (blockDim.x >> 5) + (threadIdx.x >> 5);
    int lane = threadIdx.x & 31;
    if (gw >= nb * np) return;
    int b = gw / np;
    const float* xb = xyz + (size_t)b * Nk * 3;
    float cx = nxyz[(size_t)gw * 3 + 0];
    float cy = nxyz[(size_t)gw * 3 + 1];
    float cz = nxyz[(size_t)gw * 3 + 2];
    int* out = idx + (size_t)gw * ns;
    int cnt = 0, firstIdx = -1;
    for (int base = 0; base < Nk && cnt < ns; base += 32) {
        int i = base + lane;
        bool ok = false;
        if (i < Nk) {
            float dx = xb[i * 3 + 0] - cx;
            float dy = xb[i * 3 + 1] - cy;
            float dz = xb[i * 3 + 2] - cz;
            ok = (dx * dx + dy * dy + dz * dz) <= r2;
        }
        unsigned mask = (unsigned)__ballot(ok);
        if (ok) {
            int pos = cnt + __popc(mask & ((1u << lane) - 1u));
            if (pos < ns) out[pos] = i;
        }
        if (firstIdx < 0 && mask) firstIdx = base + __ffs(mask) - 1;
        cnt += __popc(mask);
    }
    if (lane == 0) {
        int fill = (firstIdx >= 0) ? firstIdx : (Nk - 1);
        int start = (cnt < ns) ? cnt : ns;
        for (int t = start; t < ns; ++t) out[t] = fill;
    }
}

// ---------------------------------------------------------------- group (relative xyz + feats)
__global__ void group_kernel(const float* __restrict__ xyz, const float* __restrict__ feat,
                             const float* __restrict__ nxyz, const int* __restrict__ idx,
                             float* __restrict__ g, int Nk, int np, int ns, int Cf, int nb)
{
    size_t gid = (size_t)blockIdx.x * blockDim.x + threadIdx.x;
    size_t total = (size_t)nb * np * ns;
    if (gid >= total) return;
    int i = idx[gid];
    size_t pj = gid / ns;
    int b = (int)(pj / np);
    const float* xp = xyz + ((size_t)b * Nk + i) * 3;
    const float* cp = nxyz + pj * 3;
    float* o = g + gid * (size_t)(3 + Cf);
    o[0] = xp[0] - cp[0];
    o[1] = xp[1] - cp[1];
    o[2] = xp[2] - cp[2];
    const float* fb = feat + ((size_t)b * Nk + i) * Cf;
    for (int c = 0; c < Cf; ++c) o[3 + c] = fb[c];
}

// ---------------------------------------------------------------- fp32 WMMA GEMM: out = act(A[M,K] @ W[K,N] + bias)
// mode: 0 = none, 1 = relu, 2 = sigmoid
// Wave = one 16x32 C tile (two 16x16 n-tiles) via V_WMMA_F32_16X16X4_F32,
// reusing the expensive strided A fragment across both WMMAs.
// Edge policy: M/N edges use clamped (always in-bounds) addresses; garbage is
// confined to unsaved rows/columns of D. Only the K tail zero-masks the A
// fragment (register cndmask, no EXEC branches in the hot loop).
#define GEMM_WAVES 4
__global__ void gemm_kernel(const float* __restrict__ A, const float* __restrict__ W,
                            const float* __restrict__ bias, float* __restrict__ out,
                            int M, int K, int Nc, int mode)
{
    int lane  = threadIdx.x & 31;
    int wave  = threadIdx.x >> 5;
    int mtile = blockIdx.x * GEMM_WAVES + wave;
    int m0 = mtile * 16;
    int n0 = blockIdx.y * 32;
    if (m0 >= M) return;
    int half = lane >> 4;      // 0: lanes 0-15 hold K=0,1 of step; 1: lanes 16-31 hold K=2,3
    int l16  = lane & 15;
    int row  = min(m0 + l16, M - 1);
    int col0 = n0 + l16;
    int col1 = n0 + 16 + l16;
    bool cok0 = col0 < Nc;
    bool cok1 = col1 < Nc;
    int colc0 = cok0 ? col0 : (Nc - 1);
    int colc1 = cok1 ? col1 : (Nc - 1);
    int ka    = 2 * half;      // lane's k offset inside each 4-wide step

    const float* Ap  = A + (size_t)row * K + ka;     // advances by 4 per step
    const float* Bp0 = W + (size_t)ka * Nc + colc0;  // advances by 4*Nc per step
    const float* Bp1 = W + (size_t)ka * Nc + colc1;
    size_t bstep = (size_t)4 * Nc;

    v8f acc0 = {};
    v8f acc1 = {};
    int Kmain = K & ~3;
    for (int k0 = 0; k0 < Kmain; k0 += 4) {
        v2f a, b0, b1;
        a.x  = Ap[0];
        a.y  = Ap[1];
        b0.x = Bp0[0];
        b0.y = Bp0[(size_t)Nc];
        b1.x = Bp1[0];
        b1.y = Bp1[(size_t)Nc];
        acc0 = __builtin_amdgcn_wmma_f32_16x16x4_f32(false, a, false, b0, (short)0, acc0, false, false);
        acc1 = __builtin_amdgcn_wmma_f32_16x16x4_f32(false, a, false, b1, (short)0, acc1, false, false);
        Ap  += 4;
        Bp0 += bstep;
        Bp1 += bstep;
    }
    if (Kmain < K) {
        // tail: k in [Kmain, K). Clamp addresses in-bounds, zero-mask A only.
        int ia = Kmain + ka, ib = Kmain + ka + 1;
        bool oka = ia < K, okb = ib < K;
        int ca = oka ? ia : (K - 1);
        int cb = okb ? ib : (K - 1);
        const float* Ar = A + (size_t)row * K;
        float av0 = Ar[ca];
        float av1 = Ar[cb];
        v2f a, b0, b1;
        a.x  = oka ? av0 : 0.f;
        a.y  = okb ? av1 : 0.f;
        b0.x = W[(size_t)ca * Nc + colc0];
        b0.y = W[(size_t)cb * Nc + colc0];
        b1.x = W[(size_t)ca * Nc + colc1];
        b1.y = W[(size_t)cb * Nc + colc1];
        acc0 = __builtin_amdgcn_wmma_f32_16x16x4_f32(false, a, false, b0, (short)0, acc0, false, false);
        acc1 = __builtin_amdgcn_wmma_f32_16x16x4_f32(false, a, false, b1, (short)0, acc1, false, false);
    }
    if (cok0) {
        float bv = bias[col0];
        for (int r = 0; r < 8; ++r) {
            int rowD = m0 + half * 8 + r;
            if (rowD < M) {
                float v = acc0[r] + bv;
                if (mode == 1)      v = fmaxf(v, 0.f);
                else if (mode == 2) v = 1.f / (1.f + __expf(-v));
                out[(size_t)rowD * Nc + col0] = v;
            }
        }
    }
    if (cok1) {
        float bv = bias[col1];
        for (int r = 0; r < 8; ++r) {
            int rowD = m0 + half * 8 + r;
            if (rowD < M) {
                float v = acc1[r] + bv;
                if (mode == 1)      v = fmaxf(v, 0.f);
                else if (mode == 2) v = 1.f / (1.f + __expf(-v));
                out[(size_t)rowD * Nc + col1] = v;
            }
        }
    }
}

// ---------------------------------------------------------------- max-pool over samples, write into channel-offset slice
__global__ void maxpool_kernel(const float* __restrict__ g, float* __restrict__ nf,
                               int rows, int ns, int C, int stride, int coff)
{
    size_t gid = (size_t)blockIdx.x * blockDim.x + threadIdx.x;
    size_t total = (size_t)rows * C;
    if (gid >= total) return;
    int c = (int)(gid % C);
    int r = (int)(gid / C);
    const float* p = g + (size_t)r * ns * C + c;
    float m = p[0];
    for (int t = 1; t < ns; ++t) m = fmaxf(m, p[(size_t)t * C]);
    nf[(size_t)r * stride + coff + c] = m;
}

// ---------------------------------------------------------------- bilinear grid sample (ref semantics)
__global__ void grid_sample_kernel(const float* __restrict__ img, const float* __restrict__ pts,
                                   float* __restrict__ out, int H, int W, int C, int nb, int np)
{
    size_t gid = (size_t)blockIdx.x * blockDim.x + threadIdx.x;
    size_t total = (size_t)nb * np * C;
    if (gid >= total) return;
    int c = (int)(gid % C);
    size_t pj = gid / C;
    int b = (int)(pj / np);
    float gx = pts[pj * 2 + 0], gy = pts[pj * 2 + 1];
    float x = (gx + 1.f) * (W * 0.5f) - 0.5f;
    float y = (gy + 1.f) * (H * 0.5f) - 0.5f;
    float x0 = floorf(x), y0 = floorf(y);
    float wx = x - x0, wy = y - y0;
    int x0i = min(max((int)x0, 0), W - 1);
    int x1i = min(x0i + 1, W - 1);
    int y0i = min(max((int)y0, 0), H - 1);
    int y1i = min(y0i + 1, H - 1);
    const float* im = img + ((size_t)b * C + c) * H * W;
    out[gid] = im[y0i * W + x0i] * (1.f - wx) * (1.f - wy)
             + im[y0i * W + x1i] * wx * (1.f - wy)
             + im[y1i * W + x0i] * (1.f - wx) * wy
             + im[y1i * W + x1i] * wx * wy;
}

// ---------------------------------------------------------------- nf = nf * gate + proj
__global__ void combine_kernel(float* __restrict__ nf, const float* __restrict__ gate,
                               const float* __restrict__ proj, size_t n)
{
    size_t gid = (size_t)blockIdx.x * blockDim.x + threadIdx.x;
    if (gid < n) nf[gid] = nf[gid] * gate[gid] + proj[gid];
}

// ---------------------------------------------------------------- 3-NN (brute force, stable tie-break)
__global__ void three_nn_kernel(const float* __restrict__ q, const float* __restrict__ ref,
                                int* __restrict__ idx3, float* __restrict__ w3,
                                int Nq, int Nr, int nb)
{
    int r = blockIdx.x * blockDim.x + threadIdx.x;
    if (r >= nb * Nq) return;
    int b = r / Nq;
    float qx = q[(size_t)r * 3 + 0], qy = q[(size_t)r * 3 + 1], qz = q[(size_t)r * 3 + 2];
    const float* rb = ref + (size_t)b * Nr * 3;
    float d0 = 3.4e38f, d1 = 3.4e38f, d2 = 3.4e38f;
    int i0 = 0, i1 = 0, i2 = 0;
    for (int i = 0; i < Nr; ++i) {
        float dx = rb[i * 3 + 0] - qx;
        float dy = rb[i * 3 + 1] - qy;
        float dz = rb[i * 3 + 2] - qz;
        float d = dx * dx + dy * dy + dz * dz;
        if (d < d0)      { d2 = d1; i2 = i1; d1 = d0; i1 = i0; d0 = d; i0 = i; }
        else if (d < d1) { d2 = d1; i2 = i1; d1 = d; i1 = i; }
        else if (d < d2) { d2 = d; i2 = i; }
    }
    float s0 = sqrtf(fmaxf(d0, 0.f)), s1 = sqrtf(fmaxf(d1, 0.f)), s2 = sqrtf(fmaxf(d2, 0.f));
    float w0 = 1.f / (s0 + 1e-8f), w1 = 1.f / (s1 + 1e-8f), w2 = 1.f / (s2 + 1e-8f);
    float s = w0 + w1 + w2;
    w0 /= s; w1 /= s; w2 /= s;
    idx3[(size_t)r * 3 + 0] = i0; idx3[(size_t)r * 3 + 1] = i1; idx3[(size_t)r * 3 + 2] = i2;
    w3[(size_t)r * 3 + 0] = w0;   w3[(size_t)r * 3 + 1] = w1;   w3[(size_t)r * 3 + 2] = w2;
}

// ---------------------------------------------------------------- FP concat: [skip | interp(prev via 3-NN)]
__global__ void fp_concat_kernel(const float* __restrict__ skipF, const float* __restrict__ prevF,
                                 const int* __restrict__ idx3, const float* __restrict__ w3,
                                 float* __restrict__ out, int rows, int Nq, int Nr,
                                 int skipC, int Cp)
{
    int Cout = skipC + Cp;
    size_t gid = (size_t)blockIdx.x * blockDim.x + threadIdx.x;
    size_t total = (size_t)rows * Cout;
    if (gid >= total) return;
    int c = (int)(gid % Cout);
    int r = (int)(gid / Cout);
    if (c < skipC) {
        out[gid] = skipF[(size_t)r * skipC + c];
    } else {
        int cc = c - skipC;
        int b = r / Nq;
        size_t basep = (size_t)b * Nr;
        float v = 0.f;
        for (int j = 0; j < 3; ++j)
            v += prevF[(basep + idx3[(size_t)r * 3 + j]) * Cp + cc] * w3[(size_t)r * 3 + j];
        out[gid] = v;
    }
}

// ================================================================ host
extern "C" void kernel_launch(void* const* d_in, const int* in_sizes, int n_in,
                              void* d_out, int out_size, void* d_ws, size_t ws_size,
                              hipStream_t stream)
{
    (void)in_sizes; (void)n_in; (void)out_size; (void)ws_size;
    const int B = 2, N = 16384, CIMG = 256;
    const int NL[5] = {16384, 4096, 1024, 256, 64};
    static const float RAD2[4][2] = {{0.01f, 0.25f}, {0.25f, 1.f}, {1.f, 4.f}, {4.f, 16.f}};
    static const int MLPC[4][2][4] = {
        {{4, 16, 16, 32},      {4, 32, 32, 64}},
        {{99, 64, 64, 128},    {99, 64, 96, 128}},
        {{259, 128, 196, 256}, {259, 128, 196, 256}},
        {{515, 256, 256, 512}, {515, 256, 384, 512}}};
    static const int SA_OUT[4] = {96, 256, 512, 1024};
    static const int CH_IN[4]  = {1, 96, 256, 512};
    static const int FPD[4][3] = {{257, 128, 128}, {608, 256, 256}, {768, 512, 512}, {1536, 512, 512}};
    static const int SKIPC[4]  = {1, 96, 256, 512};
    static const int IMH[4] = {96, 48, 24, 12}, IMW[4] = {320, 160, 80, 40};

    const float* points = (const float*)d_in[0];
    const float* ptsimg = (const float*)d_in[1];
    auto P = [&](int i) { return (const float*)d_in[i]; };
    // params flattened in insertion order: SA levels (6 W/b per scale + 4 bu), then FP (4 W/b + 4 td)
    auto saW  = [&](int k, int s, int l) { return P(2 + 16 * k + 6 * s + 2 * l); };
    auto saB  = [&](int k, int s, int l) { return P(2 + 16 * k + 6 * s + 2 * l + 1); };
    auto buP  = [&](int k, int j) { return P(2 + 16 * k + 12 + j); };   // 0:Wg 1:bg 2:Wp 3:bp
    auto fpW  = [&](int k, int l) { return P(66 + 8 * k + 2 * l); };
    auto fpB  = [&](int k, int l) { return P(66 + 8 * k + 2 * l + 1); };
    auto tdP  = [&](int k, int j) { return P(66 + 8 * k + 4 + j); };    // 0:Wg 1:bg 2:Wp 3:bp
    auto imgBU = [&](int k) { return P(98 + k); };
    auto imgTD = [&](int k) { return P(102 + k); };

    // bump allocator over workspace
    char* base = (char*)d_ws;
    size_t off = 0;
    auto allocF = [&](size_t n) -> float* { float* p = (float*)(base + off); off += (n * sizeof(float) + 255) & ~(size_t)255; return p; };
    auto allocI = [&](size_t n) -> int*   { int* p   = (int*)(base + off);   off += (n * sizeof(int)   + 255) & ~(size_t)255; return p; };

    float* xyzL[5]; float* featL[5]; float* selL[5];
    xyzL[0]  = allocF((size_t)B * N * 3);
    featL[0] = allocF((size_t)B * N);
    selL[0]  = allocF((size_t)B * N * 2);
    for (int k = 1; k < 5; ++k) {
        xyzL[k]  = allocF((size_t)B * NL[k] * 3);
        selL[k]  = allocF((size_t)B * NL[k] * 2);
        featL[k] = allocF((size_t)B * NL[k] * SA_OUT[k - 1]);
    }
    float* dists = allocF((size_t)B * N);
    int*   fidx  = allocI((size_t)B * NL[1]);
    int*   idxbuf = allocI((size_t)B * 4096 * 32);
    float* gA = allocF((size_t)B * 4096 * 32 * 64);
    float* gB = allocF((size_t)B * 4096 * 32 * 64);
    float* samp = allocF((size_t)B * N * CIMG);
    float* gate = allocF((size_t)B * N * 128);
    float* proj = allocF((size_t)B * N * 128);
    float* gin  = allocF((size_t)B * N * 257);
    float* ftmp = allocF((size_t)B * N * 128);
    int*   idx3 = allocI((size_t)B * N * 3);
    float* w3   = allocF((size_t)B * N * 3);
    float* fpOut[4];
    for (int k = 0; k < 4; ++k)
        fpOut[k] = (k == 0) ? (float*)d_out : allocF((size_t)B * NL[k] * FPD[k][2]);

    auto gemm = [&](const float* A, const float* W, const float* bias, float* out,
                    int M, int K, int Nc, int mode) {
        int mt = (M + 15) / 16, nt = (Nc + 31) / 32;
        dim3 grid((mt + GEMM_WAVES - 1) / GEMM_WAVES, nt);
        gemm_kernel<<<grid, dim3(32 * GEMM_WAVES), 0, stream>>>(A, W, bias, out, M, K, Nc, mode);
    };

    int total0 = B * N;
    prep_kernel<<<(total0 + 255) / 256, 256, 0, stream>>>(points, ptsimg, xyzL[0], featL[0], selL[0],
                                                          total0, 1.f / 1280.f, 1.f / 384.f);

    // ---------------- set abstraction levels ----------------
    for (int k = 0; k < 4; ++k) {
        int Nk = NL[k], np = NL[k + 1];
        fps_kernel<<<B, 1024, 0, stream>>>(xyzL[k], dists, fidx, Nk, np);
        gather_kernel<<<((size_t)B * np * 3 + 255) / 256, 256, 0, stream>>>(xyzL[k], fidx, xyzL[k + 1], Nk, np, 3, B);
        gather_kernel<<<((size_t)B * np * 2 + 255) / 256, 256, 0, stream>>>(selL[k], fidx, selL[k + 1], Nk, np, 2, B);
        int coff = 0;
        for (int s = 0; s < 2; ++s) {
            int ns = (s == 0) ? 16 : 32;
            int rows = B * np * ns;
            int nwaves = B * np;
            ball_query_kernel<<<(nwaves + 3) / 4, 128, 0, stream>>>(xyzL[k], xyzL[k + 1], idxbuf,
                                                                    Nk, np, RAD2[k][s], ns, B);
            group_kernel<<<((size_t)B * np * ns + 255) / 256, 256, 0, stream>>>(
                xyzL[k], featL[k], xyzL[k + 1], idxbuf, gA, Nk, np, ns, CH_IN[k], B);
            const int* ch = MLPC[k][s];
            float* src = gA; float* dst = gB;
            for (int l = 0; l < 3; ++l) {
                gemm(src, saW(k, s, l), saB(k, s, l), dst, rows, ch[l], ch[l + 1], 1);
                float* t = src; src = dst; dst = t;
            }
            maxpool_kernel<<<((size_t)B * np * ch[3] + 255) / 256, 256, 0, stream>>>(
                src, featL[k + 1], B * np, ns, ch[3], SA_OUT[k], coff);
            coff += ch[3];
        }
        grid_sample_kernel<<<((size_t)B * np * CIMG + 255) / 256, 256, 0, stream>>>(
            imgBU(k), selL[k + 1], samp, IMH[k], IMW[k], CIMG, B, np);
        gemm(samp, buP(k, 0), buP(k, 1), gate, B * np, CIMG, SA_OUT[k], 2);
        gemm(samp, buP(k, 2), buP(k, 3), proj, B * np, CIMG, SA_OUT[k], 0);
        combine_kernel<<<((size_t)B * np * SA_OUT[k] + 255) / 256, 256, 0, stream>>>(
            featL[k + 1], gate, proj, (size_t)B * np * SA_OUT[k]);
    }

    // ---------------- feature propagation levels ----------------
    float* curFeat[5];
    for (int k = 0; k < 5; ++k) curFeat[k] = featL[k];
    for (int k = 3; k >= 0; --k) {
        int Nq = NL[k], Nr = NL[k + 1], rows = B * Nq;
        three_nn_kernel<<<(rows + 255) / 256, 256, 0, stream>>>(xyzL[k], xyzL[k + 1], idx3, w3, Nq, Nr, B);
        int Cp = (k == 3) ? SA_OUT[3] : FPD[k + 1][2];
        fp_concat_kernel<<<((size_t)rows * FPD[k][0] + 255) / 256, 256, 0, stream>>>(
            curFeat[k], curFeat[k + 1], idx3, w3, gin, rows, Nq, Nr, SKIPC[k], Cp);
        gemm(gin,  fpW(k, 0), fpB(k, 0), ftmp,     rows, FPD[k][0], FPD[k][1], 1);
        gemm(ftmp, fpW(k, 1), fpB(k, 1), fpOut[k], rows, FPD[k][1], FPD[k][2], 1);
        grid_sample_kernel<<<((size_t)rows * CIMG + 255) / 256, 256, 0, stream>>>(
            imgTD(k), selL[k], samp, IMH[k], IMW[k], CIMG, B, Nq);
        gemm(samp, tdP(k, 0), tdP(k, 1), gate, rows, CIMG, FPD[k][2], 2);
        gemm(samp, tdP(k, 2), tdP(k, 3), proj, rows, CIMG, FPD[k][2], 0);
        combine_kernel<<<((size_t)rows * FPD[k][2] + 255) / 256, 256, 0, stream>>>(
            fpOut[k], gate, proj, (size_t)rows * FPD[k][2]);
        curFeat[k] = fpOut[k];
    }
}